// AttnDecoderRNN_85315230368031
// MI455X (gfx1250) — compile-verified
//
#include <hip/hip_runtime.h>
#include <hip/hip_bf16.h>
#include <math.h>

// Problem constants (from reference): V=50000, H=1024, L=512, B=8
#define VV 50000
#define HH 1024
#define LL 512
#define BB 8

typedef __attribute__((ext_vector_type(2))) float v2f;
typedef __attribute__((ext_vector_type(8))) float v8f;

// ---------------- workspace layout (floats) ----------------
#define WS_XH      0         // B x 2H
#define WS_AW      16384     // B x L
#define WS_APPLIED 20480     // B x H
#define WS_G       28672     // B x H
#define WS_GI      36864     // B x 3H
#define WS_GH      61440     // B x 3H
#define WS_HNEW    86016     // B x H
#define WS_APAD    94208     // 16 x H
#define WS_LOGITS  110592    // B x V

__device__ __forceinline__ float sigmoidf_(float x) { return 1.0f / (1.0f + expf(-x)); }

// ---------------- 1) xh = concat(emb[input], hidden) ----------------
__global__ void k_xh(const int* __restrict__ input, const float* __restrict__ hidden,
                     const float* __restrict__ emb, float* __restrict__ xh) {
  int idx = blockIdx.x * blockDim.x + threadIdx.x;            // B*2H threads
  if (idx >= BB * 2 * HH) return;
  int b = idx / (2 * HH);
  int j = idx % (2 * HH);
  float v;
  if (j < HH) v = emb[(size_t)input[b] * HH + j];
  else        v = hidden[b * HH + (j - HH)];
  xh[idx] = v;
}

// ---------------- 2) attention scores: aw[b,l] = xh[b] . attn_w[l] + attn_b[l] ----------
__global__ void k_attn_scores(const float* __restrict__ xh, const float* __restrict__ attn_w,
                              const float* __restrict__ attn_b, float* __restrict__ aw) {
  int wid  = (blockIdx.x * blockDim.x + threadIdx.x) >> 5;    // one wave per output
  int lane = threadIdx.x & 31;
  if (wid >= BB * LL) return;
  int b = wid / LL, l = wid % LL;
  const float* xr = xh + b * 2 * HH;
  const float* wr = attn_w + (size_t)l * 2 * HH;
  float acc = 0.f;
  for (int i = lane; i < 2 * HH; i += 32) acc += xr[i] * wr[i];
  for (int o = 16; o > 0; o >>= 1) acc += __shfl_down(acc, o, 32);
  if (lane == 0) aw[b * LL + l] = acc + attn_b[l];
}

// ---------------- 3) softmax over L (in place on aw) ----------------
__global__ void k_softmax_L(float* __restrict__ aw) {
  __shared__ float sm[LL];
  int b = blockIdx.x, t = threadIdx.x;                        // blockDim = 512
  float v = aw[b * LL + t];
  sm[t] = v; __syncthreads();
  for (int s = LL / 2; s > 0; s >>= 1) { if (t < s) sm[t] = fmaxf(sm[t], sm[t + s]); __syncthreads(); }
  float mx = sm[0]; __syncthreads();
  float e = expf(v - mx);
  sm[t] = e; __syncthreads();
  for (int s = LL / 2; s > 0; s >>= 1) { if (t < s) sm[t] += sm[t + s]; __syncthreads(); }
  aw[b * LL + t] = e / sm[0];
}

// ---------------- 4) applied[b,h] = sum_l aw[b,l] * enc[b,l,h] ----------------
__global__ void k_applied(const float* __restrict__ aw, const float* __restrict__ enc,
                          float* __restrict__ applied) {
  __shared__ float s_aw[LL];
  int b = blockIdx.x, h = threadIdx.x;                        // blockDim = 1024
  if (threadIdx.x < LL) s_aw[threadIdx.x] = aw[b * LL + threadIdx.x];
  __syncthreads();
  float acc = 0.f;
  const float* e = enc + (size_t)b * LL * HH + h;
  for (int l = 0; l < LL; ++l) acc += s_aw[l] * e[(size_t)l * HH];
  applied[b * HH + h] = acc;
}

// ---------------- 5) g[b,h] = relu(concat(x, applied)[b] . comb_w[h] + comb_b[h]) -------
__global__ void k_combine(const float* __restrict__ xh, const float* __restrict__ applied,
                          const float* __restrict__ comb_w, const float* __restrict__ comb_b,
                          float* __restrict__ g) {
  int wid  = (blockIdx.x * blockDim.x + threadIdx.x) >> 5;    // one wave per output, B*H outputs
  int lane = threadIdx.x & 31;
  if (wid >= BB * HH) return;
  int b = wid / HH, ho = wid % HH;
  const float* wr = comb_w + (size_t)ho * 2 * HH;
  const float* x  = xh + b * 2 * HH;            // first H entries are x
  const float* ap = applied + b * HH;
  float acc = 0.f;
  for (int i = lane; i < HH; i += 32)     acc += x[i]  * wr[i];
  for (int i = lane; i < HH; i += 32)     acc += ap[i] * wr[HH + i];
  for (int o = 16; o > 0; o >>= 1) acc += __shfl_down(acc, o, 32);
  if (lane == 0) g[b * HH + ho] = fmaxf(acc + comb_b[ho], 0.f);
}

// ---------------- 6) generic dot: out[b,n] = A[b] . W[n] + bias[n]  (K = H) -------------
__global__ void k_dot(const float* __restrict__ A, const float* __restrict__ W,
                      const float* __restrict__ bias, float* __restrict__ out, int N) {
  int wid  = (blockIdx.x * blockDim.x + threadIdx.x) >> 5;    // one wave per output, B*N outputs
  int lane = threadIdx.x & 31;
  if (wid >= BB * N) return;
  int b = wid / N, n = wid % N;
  const float* ar = A + (size_t)b * HH;
  const float* wr = W + (size_t)n * HH;
  float acc = 0.f;
  for (int i = lane; i < HH; i += 32) acc += ar[i] * wr[i];
  for (int o = 16; o > 0; o >>= 1) acc += __shfl_down(acc, o, 32);
  if (lane == 0) out[b * N + n] = acc + bias[n];
}

// ---------------- 7) GRU gates + h_new ----------------
__global__ void k_gru(const float* __restrict__ gi, const float* __restrict__ gh,
                      const float* __restrict__ hidden, float* __restrict__ hnew,
                      float* __restrict__ out_tail) {
  int idx = blockIdx.x * blockDim.x + threadIdx.x;            // B*H threads
  if (idx >= BB * HH) return;
  int b = idx / HH, h = idx % HH;
  const float* gib = gi + b * 3 * HH;
  const float* ghb = gh + b * 3 * HH;
  float ir = gib[h], iz = gib[HH + h], inn = gib[2 * HH + h];
  float hr = ghb[h], hz = ghb[HH + h], hn  = ghb[2 * HH + h];
  float r = sigmoidf_(ir + hr);
  float z = sigmoidf_(iz + hz);
  float n = tanhf(inn + r * hn);
  float hprev = hidden[b * HH + h];
  float hv = (1.0f - z) * n + z * hprev;
  hnew[idx] = hv;
  out_tail[idx] = hv;                                         // h_new part of d_out
}

// ---------------- 8) zero-pad h_new to 16 rows for WMMA ----------------
__global__ void k_padA(const float* __restrict__ hnew, float* __restrict__ Apad) {
  int idx = blockIdx.x * blockDim.x + threadIdx.x;            // 16*H threads
  if (idx >= 16 * HH) return;
  int m = idx / HH;
  Apad[idx] = (m < BB) ? hnew[idx] : 0.0f;
}

// ---------------- 9) logits = Apad(16xH) x out_w^T (HxV) + out_b  via f32 WMMA ----------
// One wave per 16-column vocab tile. A is staged per-block in LDS (shared by 8 waves) as
// per-kstep rows of 16 float4 (A[m][4k..4k+3]) with a 68-float padded stride so the
// ds_load_b64 fragment reads hit all 64 LDS banks conflict-free. Inner loop per K-step:
//   ds_load_b64 (A frag) + global_load_b64 (W frag) + v_wmma_f32_16x16x4_f32.
// K is staged in two 512-wide chunks (34.8 KB LDS).
#define KCH 128                         // k-steps per chunk (512 K per chunk, 2 chunks)
__global__ void k_logits_wmma(const float* __restrict__ Apad, const float* __restrict__ Wout,
                              const float* __restrict__ outb, float* __restrict__ logits) {
  __shared__ float As[KCH * 68];        // 128 ksteps x (16 float4 + 1 float4 pad)
  const int lane = threadIdx.x & 31;
  const int wave = threadIdx.x >> 5;
  const int tile = blockIdx.x * 8 + wave;                     // V/16 = 3125 tiles
  const bool active = (tile < VV / 16);
  const int v0 = active ? tile * 16 : 0;                      // tail waves compute tile 0, skip store
  const int n  = lane & 15;
  const int m  = lane & 15;
  const int hi = lane >> 4;                                   // 0 or 1 -> K offset 0/2
  const float* wrow = Wout + (size_t)(v0 + n) * HH + 2 * hi;
  const float* afr  = As + m * 4 + hi * 2;
  v8f acc = {};
  for (int chunk = 0; chunk < 2; ++chunk) {
    __syncthreads();                    // protect previous chunk's LDS reads
    // cooperative stage: 16 rows x 128 float4s, coalesced b128 reads
    for (int f = threadIdx.x; f < 16 * KCH; f += 256) {
      int mm = f >> 7;                  // 0..15
      int ks = f & (KCH - 1);           // 0..127
      float4 val = *(const float4*)(Apad + mm * HH + chunk * (4 * KCH) + ks * 4);
      *(float4*)(As + ks * 68 + mm * 4) = val;
    }
    __syncthreads();
    const float* wch = wrow + chunk * (4 * KCH);
    #pragma unroll 4
    for (int ks = 0; ks < KCH; ++ks) {
      v2f a = *(const v2f*)(afr + ks * 68);        // ds_load_b64, bank-conflict-free
      v2f b = *(const v2f*)(wch + ks * 4);         // global_load_b64, streams out_w once
      acc = __builtin_amdgcn_wmma_f32_16x16x4_f32(false, a, false, b, (short)0, acc,
                                                  false, false);
    }
  }
  if (!active) return;
  const float bias = outb[v0 + n];
  const int mbase = (lane < 16) ? 0 : 8;           // D: VGPR r holds M=r / M=8+r
  #pragma unroll
  for (int r = 0; r < 8; ++r) {
    int mm = mbase + r;
    if (mm < BB) logits[(size_t)mm * VV + v0 + n] = acc[r] + bias;
  }
}

// ---------------- 10) log_softmax over V per row ----------------
__global__ void k_logsoftmax(const float* __restrict__ logits, float* __restrict__ out) {
  __shared__ float red[1024];
  int b = blockIdx.x, t = threadIdx.x;                        // blockDim = 1024
  const float* row = logits + (size_t)b * VV;
  float mx = -INFINITY;
  for (int v = t; v < VV; v += 1024) mx = fmaxf(mx, row[v]);
  red[t] = mx; __syncthreads();
  for (int s = 512; s > 0; s >>= 1) { if (t < s) red[t] = fmaxf(red[t], red[t + s]); __syncthreads(); }
  mx = red[0]; __syncthreads();
  float sum = 0.f;
  for (int v = t; v < VV; v += 1024) sum += expf(row[v] - mx);
  red[t] = sum; __syncthreads();
  for (int s = 512; s > 0; s >>= 1) { if (t < s) red[t] += red[t + s]; __syncthreads(); }
  float lse = logf(red[0]);
  float* orow = out + (size_t)b * VV;
  for (int v = t; v < VV; v += 1024) orow[v] = row[v] - mx - lse;
}

extern "C" void kernel_launch(void* const* d_in, const int* in_sizes, int n_in,
                              void* d_out, int out_size, void* d_ws, size_t ws_size,
                              hipStream_t stream) {
  const int*   input  = (const int*)  d_in[0];
  const float* hidden = (const float*)d_in[1];
  const float* enc    = (const float*)d_in[2];
  const float* emb    = (const float*)d_in[3];
  const float* attn_w = (const float*)d_in[4];
  const float* attn_b = (const float*)d_in[5];
  const float* comb_w = (const float*)d_in[6];
  const float* comb_b = (const float*)d_in[7];
  const float* w_ih   = (const float*)d_in[8];
  const float* w_hh   = (const float*)d_in[9];
  const float* b_ih   = (const float*)d_in[10];
  const float* b_hh   = (const float*)d_in[11];
  const float* out_w  = (const float*)d_in[12];
  const float* out_b  = (const float*)d_in[13];

  float* ws      = (float*)d_ws;
  float* xh      = ws + WS_XH;
  float* aw      = ws + WS_AW;
  float* applied = ws + WS_APPLIED;
  float* g       = ws + WS_G;
  float* gi      = ws + WS_GI;
  float* gh      = ws + WS_GH;
  float* hnew    = ws + WS_HNEW;
  float* Apad    = ws + WS_APAD;
  float* logits  = ws + WS_LOGITS;

  float* out_logp = (float*)d_out;                 // B*V
  float* out_h    = out_logp + (size_t)BB * VV;    // B*H

  // 1) gather + concat
  k_xh<<<(BB * 2 * HH + 255) / 256, 256, 0, stream>>>(input, hidden, emb, xh);
  // 2) attention scores (one wave per output)
  k_attn_scores<<<(BB * LL * 32 + 255) / 256, 256, 0, stream>>>(xh, attn_w, attn_b, aw);
  // 3) softmax over L
  k_softmax_L<<<BB, LL, 0, stream>>>(aw);
  // 4) attention apply
  k_applied<<<BB, HH, 0, stream>>>(aw, enc, applied);
  // 5) combine + relu
  k_combine<<<(BB * HH * 32 + 255) / 256, 256, 0, stream>>>(xh, applied, comb_w, comb_b, g);
  // 6) GRU input/hidden GEMMs
  k_dot<<<(BB * 3 * HH * 32 + 255) / 256, 256, 0, stream>>>(g,      w_ih, b_ih, gi, 3 * HH);
  k_dot<<<(BB * 3 * HH * 32 + 255) / 256, 256, 0, stream>>>(hidden, w_hh, b_hh, gh, 3 * HH);
  // 7) gates + h_new (also writes h_new output)
  k_gru<<<(BB * HH + 255) / 256, 256, 0, stream>>>(gi, gh, hidden, hnew, out_h);
  // 8) pad A to 16 rows
  k_padA<<<(16 * HH + 255) / 256, 256, 0, stream>>>(hnew, Apad);
  // 9) big vocab GEMM via f32 WMMA: 3125 tiles, 8 waves/block, LDS-staged A
  k_logits_wmma<<<(VV / 16 + 7) / 8, 256, 0, stream>>>(Apad, out_w, out_b, logits);
  // 10) log_softmax rows
  k_logsoftmax<<<BB, 1024, 0, stream>>>(logits, out_logp);
}